// QuadraticNetCholesky_18674517803680
// MI455X (gfx1250) — compile-verified
//
#include <hip/hip_runtime.h>

// QuadraticNetCholesky for MI455X (gfx1250, wave32).
//
// Per wave: 16 samples. GEMMs on the matrix pipe with V_WMMA_F32_16X16X4_F32
// (fp32 in, fp32 acc — matches the fp32 reference), activations on VALU,
// intermediate activations staged in padded LDS, L*L^T fully unrolled in
// registers (2 lanes per sample), b128 coalesced output stores.
//
// WMMA f32 16x16x4 layouts (CDNA5 ISA):
//   A (16x4): lane<16 -> row=lane,   K={0,1}; lane>=16 -> row=lane-16, K={2,3}
//   B (4x16): lane<16 -> col=lane,   K={0,1}; lane>=16 -> col=lane-16, K={2,3}
//   C (16x16): vgpr r, lane<16 -> (row=r, col=lane); lane>=16 -> (row=r+8, col=lane-16)

typedef float v2f __attribute__((ext_vector_type(2)));
typedef float v8f __attribute__((ext_vector_type(8)));

#define H1S 132  // h1 row stride (floats), 128 padded (bank-conflict break, keeps 8B align)
#define H2S 36   // h2 row stride, 32 padded
#define VBS 84   // v  row stride, 80 padded (16B aligned rows: 84*4 = 336 = 21*16)

__device__ __forceinline__ float elu_f(float t) {
  return t > 0.0f ? t : (__expf(t) - 1.0f);
}
__device__ __forceinline__ float softplus_f(float t) {
  return __logf(1.0f + __expf(t));
}

// Rows i = P, P+2, ..., P+10 of M = L*L^T for one sample; all indices
// compile-time so Lf[] stays in VGPRs (no scratch).
template <int P>
__device__ __forceinline__ void llt_rows(const float (&Lf)[80], float* __restrict__ outp) {
#pragma unroll
  for (int ii = 0; ii < 6; ++ii) {
    const int i = 2 * ii + P;
    float row[12];
#pragma unroll
    for (int k = 0; k < 12; ++k) {
      float a = 0.0f;
#pragma unroll
      for (int j = 0; j < 12; ++j) {
        if (j <= i && j <= k)  // folds: i,k,j constant after unroll
          a = fmaf(Lf[i * (i + 1) / 2 + j], Lf[k * (k + 1) / 2 + j], a);
      }
      row[k] = a;
    }
    float4* o = (float4*)(outp + i * 12);  // 48B row offset -> 16B aligned
    o[0] = make_float4(row[0], row[1], row[2], row[3]);
    o[1] = make_float4(row[4], row[5], row[6], row[7]);
    o[2] = make_float4(row[8], row[9], row[10], row[11]);
  }
}

__global__ __launch_bounds__(128) void qnet_chol_kernel(
    const float* __restrict__ x,
    const float* __restrict__ W1, const float* __restrict__ b1,
    const float* __restrict__ W2, const float* __restrict__ b2,
    const float* __restrict__ W3, const float* __restrict__ b3,
    float* __restrict__ out) {
  __shared__ float hs1[4][16 * H1S];  // per-wave h1 (later reused for v)
  __shared__ float hs2[4][16 * H2S];  // per-wave h2

  const int lane   = threadIdx.x & 31;
  const int wave   = threadIdx.x >> 5;
  const int half   = lane >> 4;   // selects K pair {0,1} vs {2,3}
  const int lane16 = lane & 15;
  const long sbase = ((long)blockIdx.x * 4 + wave) * 16;

  float* h1 = hs1[wave];
  float* h2 = hs2[wave];
  float* vb = hs1[wave];  // v reuses h1 space (h1 dead after layer 2)

  // ---- A fragments of x: 16 samples x 12, three K-chunks of 4 ----
  v2f ax[3];
  {
    const float* xr = x + (sbase + lane16) * 12;
#pragma unroll
    for (int c = 0; c < 3; ++c)
      ax[c] = *(const v2f*)(xr + c * 4 + half * 2);  // 8B-aligned b64 loads
  }

  // ---- layer 1: h1 = elu(x @ W1^T + b1), [16x12]*[12x128], 8 N-tiles ----
#pragma unroll
  for (int nt = 0; nt < 8; ++nt) {
    const int n = nt * 16 + lane16;
    v8f acc = {};
#pragma unroll
    for (int c = 0; c < 3; ++c) {
      // B(k,n) = W1[n][k]; two contiguous k -> b64 load, L2-hot weights
      v2f bf = *(const v2f*)(W1 + n * 12 + c * 4 + half * 2);
      acc = __builtin_amdgcn_wmma_f32_16x16x4_f32(false, ax[c], false, bf,
                                                  (short)0, acc, false, false);
    }
    const float bias = b1[n];  // bias depends on column only
#pragma unroll
    for (int r = 0; r < 8; ++r)
      h1[(r + half * 8) * H1S + n] = elu_f(acc[r] + bias);
  }
  __syncthreads();

  // ---- layer 2: h2 = elu(h1 @ W2^T + b2), [16x128]*[128x32], 2 N-tiles ----
#pragma unroll
  for (int nt = 0; nt < 2; ++nt) {
    const int n = nt * 16 + lane16;
    v8f acc = {};
#pragma unroll
    for (int c = 0; c < 32; ++c) {
      v2f af = *(const v2f*)(h1 + lane16 * H1S + c * 4 + half * 2);  // ds_load_b64
      v2f bf = *(const v2f*)(W2 + n * 128 + c * 4 + half * 2);
      acc = __builtin_amdgcn_wmma_f32_16x16x4_f32(false, af, false, bf,
                                                  (short)0, acc, false, false);
    }
    const float bias = b2[n];
#pragma unroll
    for (int r = 0; r < 8; ++r)
      h2[(r + half * 8) * H2S + n] = elu_f(acc[r] + bias);
  }
  __syncthreads();

  // ---- layer 3: v = softplus(h2 @ W3^T + b3), N padded 78 -> 80, 5 N-tiles ----
#pragma unroll
  for (int nt = 0; nt < 5; ++nt) {
    const int n = nt * 16 + lane16;
    const bool valid = (n < 78);
    v8f acc = {};
#pragma unroll
    for (int c = 0; c < 8; ++c) {
      v2f af = *(const v2f*)(h2 + lane16 * H2S + c * 4 + half * 2);
      v2f bf = {0.0f, 0.0f};
      if (valid)  // divergent only around the load; WMMA runs with full EXEC
        bf = *(const v2f*)(W3 + n * 32 + c * 4 + half * 2);
      acc = __builtin_amdgcn_wmma_f32_16x16x4_f32(false, af, false, bf,
                                                  (short)0, acc, false, false);
    }
    const float bias = valid ? b3[n] : 0.0f;
#pragma unroll
    for (int r = 0; r < 8; ++r)
      vb[(r + half * 8) * VBS + n] = softplus_f(acc[r] + bias);  // cols>=78 never read
  }
  __syncthreads();

  // ---- M = L * L^T per sample; 2 lanes/sample, parity-split rows ----
  const int s = lane >> 1;   // sample 0..15 within the wave's tile
  const int p = lane & 1;    // row parity handled by this lane
  float Lf[80];
  {
    const float4* src = (const float4*)(vb + s * VBS);  // 16B-aligned ds_load_b128
#pragma unroll
    for (int t = 0; t < 20; ++t) {
      float4 q = src[t];
      Lf[4 * t + 0] = q.x; Lf[4 * t + 1] = q.y;
      Lf[4 * t + 2] = q.z; Lf[4 * t + 3] = q.w;
    }
  }
  float* outp = out + (sbase + s) * 144;
  if (p == 0) llt_rows<0>(Lf, outp);
  else        llt_rows<1>(Lf, outp);
}

extern "C" void kernel_launch(void* const* d_in, const int* in_sizes, int n_in,
                              void* d_out, int out_size, void* d_ws, size_t ws_size,
                              hipStream_t stream) {
  (void)n_in; (void)d_ws; (void)ws_size; (void)out_size;
  const float* x  = (const float*)d_in[0];
  const float* W1 = (const float*)d_in[1];
  const float* b1 = (const float*)d_in[2];
  const float* W2 = (const float*)d_in[3];
  const float* b2 = (const float*)d_in[4];
  const float* W3 = (const float*)d_in[5];
  const float* b3 = (const float*)d_in[6];
  float* out = (float*)d_out;

  const int batch = in_sizes[0] / 12;   // 262144
  const int grid  = batch / 64;         // 64 samples per block (4 waves x 16) -> 4096
  qnet_chol_kernel<<<grid, 128, 0, stream>>>(x, W1, b1, W2, b2, W3, b3, out);
}